// Loss_81698867904992
// MI455X (gfx1250) — compile-verified
//
#include <hip/hip_runtime.h>
#include <hip/hip_bf16.h>

// YOLO loss on MI455X (gfx1250): bandwidth-bound streaming reduction.
// - Conditional class reads (only ~5% of anchors are obj) to cut HBM traffic.
// - Wave-level reduction done exactly with V_WMMA_F32_16X16X4_F32 (B = ones).
// - Deterministic two-stage reduction through d_ws (no float atomics).

typedef __attribute__((ext_vector_type(2))) float v2f;
typedef __attribute__((ext_vector_type(8))) float v8f;

#define B_N   32
#define H_N   64
#define W_N   64
#define A_N   5
#define CLS_N 80
#define D_N   85                       // 5 + CLS_N
#define TOTAL (B_N * H_N * W_N * A_N)  // 655360
#define TPB   256
#define NBLK  640                      // 640*256 = 163840 threads -> 4 anchors/thread

// Exact 32-lane sum using one f32 WMMA:
//   A[m,0] = p_m (lanes 0-15), A[m,2] = p_{m+16} (lanes 16-31), A[*,1]=A[*,3]=0
//   B = ones(4x16)  =>  D[m][n] = p_m + p_{m+16}
// Lane L holds rows {r + 8*(L>=16)} for column L%16; summing its 8 D values
// gives the half-sum, and shfl_xor(16) completes the full wave sum.
__device__ __forceinline__ float wave_sum_wmma(float p) {
    v2f a; a[0] = p;    a[1] = 0.0f;
    v2f b; b[0] = 1.0f; b[1] = 1.0f;
    v8f c = {};
    v8f d = __builtin_amdgcn_wmma_f32_16x16x4_f32(
        /*neg_a=*/false, a, /*neg_b=*/false, b,
        /*c_mod=*/(short)0, c, /*reuse_a=*/false, /*reuse_b=*/false);
    float s = 0.0f;
#pragma unroll
    for (int k = 0; k < 8; ++k) s += d[k];
    s += __shfl_xor(s, 16, 32);
    return s;  // full wave sum in every lane
}

// Deterministic block sum; result valid in thread 0.
__device__ __forceinline__ float block_sum(float p) {
    __shared__ float wsum[TPB / 32];
    float s = wave_sum_wmma(p);
    const int lane = threadIdx.x & 31;
    const int wid  = threadIdx.x >> 5;
    if (lane == 0) wsum[wid] = s;
    __syncthreads();
    float tot = 0.0f;
    if (threadIdx.x == 0) {
#pragma unroll
        for (int i = 0; i < TPB / 32; ++i) tot += wsum[i];  // fixed order
    }
    return tot;
}

__device__ __forceinline__ float smooth_l1(float x) {
    float ax = fabsf(x);
    return (ax < 1.0f) ? 0.5f * x * x : ax - 0.5f;
}

__global__ void yolo_loss_partials(const float* __restrict__ out,
                                   const float* __restrict__ lab,
                                   const float* __restrict__ img_sizes,
                                   const float* __restrict__ anchors,
                                   float* __restrict__ partial) {
    const float lamda_coord = 5.0f, lamda_noobj = 0.5f;
    const float lamda_obj = 1.0f,  lamda_class = 1.0f;

    float acc = 0.0f;
    const int tid      = blockIdx.x * blockDim.x + threadIdx.x;
    const int nthreads = gridDim.x * blockDim.x;

    for (int i = tid; i < TOTAL; i += nthreads) {
        const float* o = out + (size_t)i * D_N;
        const float* l = lab + (size_t)i * D_N;

        const float l4 = l[4];
        const float o4 = o[4];
        const bool  is_obj   = (l4 >= 1.0f);
        const bool  is_noobj = (l4 == 0.0f);

        if (is_obj) {
            const int a    = i % A_N;
            const int cell = i / A_N;
            const int x    = cell % W_N;
            const int rem  = cell / W_N;
            const int y    = rem % H_N;
            const int bb   = rem / H_N;

            const float o0 = o[0], o1 = o[1], o2 = o[2], o3 = o[3];
            const float l0 = l[0], l1 = l[1], l2 = l[2], l3 = l[3];

            const float ylen = img_sizes[bb * 2 + 0] * (1.0f / (float)H_N);
            const float xlen = img_sizes[bb * 2 + 1] * (1.0f / (float)W_N);
            const float ah   = anchors[a * 2 + 0];
            const float aw   = anchors[a * 2 + 1];
            const float xg = (float)x, yg = (float)y;

            const float pxc = (xg + o0) * xlen;
            const float pyc = (yg + o1) * ylen;
            const float pw  = expf(o2) * aw;
            const float ph  = expf(o3) * ah;
            const float pxmin = pxc - 0.5f * pw, pxmax = pxc + 0.5f * pw;
            const float pymin = pyc - 0.5f * ph;
            const float pymax = pxc + 0.5f * ph;  // NOTE: reference uses pxc here

            const float txc = (xg + l0) * xlen;
            const float tyc = (yg + l1) * ylen;
            const float tw  = expf(l2) * aw;
            const float th  = expf(l3) * ah;
            const float txmin = txc - 0.5f * tw, txmax = txc + 0.5f * tw;
            const float tymin = tyc - 0.5f * th, tymax = tyc + 0.5f * th;

            const float iw = fmaxf(fminf(pxmax, txmax) - fmaxf(pxmin, txmin), 0.0f);
            const float ih = fmaxf(fminf(pymax, tymax) - fmaxf(pymin, tymin), 0.0f);
            const float inter  = iw * ih;
            const float area_p = (pxmax - pxmin) * (pymax - pymin);
            const float area_t = (txmax - txmin) * (tymax - tymin);
            const float uni    = area_p + area_t - inter;
            const float iou    = inter / uni;
            const float new_conf = l4 * iou;

            const float d0 = o0 - l0, d1 = o1 - l1;
            const float p1 = d0 * d0 + d1 * d1;
            const float p2 = smooth_l1(o2 - l2) + smooth_l1(o3 - l3);
            const float dc = o4 - new_conf;
            const float p3 = dc * dc;

            float p5 = 0.0f;
#pragma unroll 4
            for (int cch = 0; cch < CLS_N; ++cch) {
                const float dd = o[5 + cch] - l[5 + cch];
                p5 += dd * dd;
            }
            acc += lamda_coord * (p1 + p2) + lamda_obj * p3 + lamda_class * p5;
        } else if (is_noobj) {
            // iou masked to 0 where !obj  =>  new_conf = 0
            acc += lamda_noobj * o4 * o4;
        }
        // 0 < l4 < 1: contributes nothing (objf = noobjf = 0)
    }

    const float bs = block_sum(acc);  // EXEC all-ones here (full waves, reconverged)
    if (threadIdx.x == 0) partial[blockIdx.x] = bs;
}

__global__ void yolo_loss_finalize(const float* __restrict__ partial, int n,
                                   float* __restrict__ result) {
    float acc = 0.0f;
    for (int i = threadIdx.x; i < n; i += blockDim.x) acc += partial[i];
    const float bs = block_sum(acc);
    if (threadIdx.x == 0) result[0] = bs;
}

extern "C" void kernel_launch(void* const* d_in, const int* in_sizes, int n_in,
                              void* d_out, int out_size, void* d_ws, size_t ws_size,
                              hipStream_t stream) {
    const float* model_output = (const float*)d_in[0];  // (B,H,W,A*D) f32
    const float* label        = (const float*)d_in[1];  // (B,H,W,A*D) f32
    const float* img_sizes    = (const float*)d_in[2];  // (B,2) f32
    const float* anchor_boxes = (const float*)d_in[3];  // (A,2) f32
    float* result  = (float*)d_out;                     // scalar f32
    float* partial = (float*)d_ws;                      // NBLK floats of scratch

    yolo_loss_partials<<<NBLK, TPB, 0, stream>>>(model_output, label, img_sizes,
                                                 anchor_boxes, partial);
    yolo_loss_finalize<<<1, TPB, 0, stream>>>(partial, NBLK, result);
}